// CenterLoss_11038065951186
// MI455X (gfx1250) — compile-verified
//
#include <hip/hip_runtime.h>

typedef __attribute__((ext_vector_type(2))) float v2f;
typedef __attribute__((ext_vector_type(8))) float v8f;

#define D_DIM 512
#define ROWS_PER_BLOCK 16
#define LDS_STRIDE 17            // pad 16 -> 17 to kill LDS bank conflicts
#define CL_MIN 1e-12f
#define CL_MAX 1e12f

// Stage 1: each block handles 16 rows. 16 threads per row do coalesced
// float4 streaming of feature row + gathered center row, accumulating
// sum((f-c)^2) partials. Partials are reduced across the 16 lanes of each
// row with V_WMMA_F32_16X16X4_F32 using an all-ones A matrix:
//   D[m][n] = sum_k A[m][k]*B[k][n] = sum_k B[k][n]
// Four accumulating WMMAs over the 16 partials per row give the full
// row sums in every lane's acc[0] (all D rows identical with ones-A).
__global__ __launch_bounds__(256) void center_loss_stage1(
    const float* __restrict__ features,
    const long long* __restrict__ labels,
    const float* __restrict__ centers,
    float* __restrict__ partials)
{
    __shared__ float lds_p[ROWS_PER_BLOCK * LDS_STRIDE];

    const int tid = threadIdx.x;
    const int row = tid >> 4;   // 0..15 row within block
    const int sub = tid & 15;   // 0..15 lane-group within row
    const long long grow = (long long)blockIdx.x * ROWS_PER_BLOCK + row;

    const long long cls = labels[2 * grow + 1];   // labels[:, -1], int64
    const float* __restrict__ frow = features + grow * D_DIM;
    const float* __restrict__ crow = centers + cls * D_DIM;

    float s = 0.0f;
#pragma unroll
    for (int j = 0; j < D_DIM / 64; ++j) {
        const int c0 = sub * 4 + j * 64;          // coalesced 256B per 16 lanes
        float4 f = *(const float4*)(frow + c0);
        float4 c = *(const float4*)(crow + c0);
        float dx = f.x - c.x;
        float dy = f.y - c.y;
        float dz = f.z - c.z;
        float dw = f.w - c.w;
        s = fmaf(dx, dx, s);
        s = fmaf(dy, dy, s);
        s = fmaf(dz, dz, s);
        s = fmaf(dw, dw, s);
    }
    lds_p[row * LDS_STRIDE + sub] = s;
    __syncthreads();

    if (tid < 32) {                 // wave 0 only (uniform branch: EXEC all-1s)
        const int lane = tid;
        const int n  = lane & 15;   // B/C/D column index = lane % 16
        const int hi = lane >> 4;   // half-wave selects K pair {0,1} vs {2,3}

        v2f a; a[0] = 1.0f; a[1] = 1.0f;   // ones A-matrix
        v8f acc = {};
#pragma unroll
        for (int j = 0; j < 4; ++j) {
            v2f b;
            b[0] = lds_p[n * LDS_STRIDE + 4 * j + 2 * hi + 0];
            b[1] = lds_p[n * LDS_STRIDE + 4 * j + 2 * hi + 1];
            acc = __builtin_amdgcn_wmma_f32_16x16x4_f32(
                false, a, false, b, (short)0, acc, false, false);
        }
        // acc[0] == row-sum of partials for row n (every D row identical)
        float d2 = acc[0];
        d2 = fminf(fmaxf(d2, CL_MIN), CL_MAX);

        float v = (hi == 0) ? d2 : 0.0f;    // count each row exactly once
#pragma unroll
        for (int m = 16; m >= 1; m >>= 1)
            v += __shfl_xor(v, m, 32);
        if (lane == 0)
            partials[blockIdx.x] = v;
    }
}

// Stage 2: deterministic fixed-order tree reduction of per-block partials.
__global__ __launch_bounds__(256) void center_loss_stage2(
    const float* __restrict__ partials,
    float* __restrict__ out,
    int nparts, float inv_n)
{
    __shared__ float red[256];
    float s = 0.0f;
    for (int i = threadIdx.x; i < nparts; i += 256)
        s += partials[i];
    red[threadIdx.x] = s;
    __syncthreads();
#pragma unroll
    for (int off = 128; off >= 1; off >>= 1) {
        if ((int)threadIdx.x < off)
            red[threadIdx.x] += red[threadIdx.x + off];
        __syncthreads();
    }
    if (threadIdx.x == 0)
        out[0] = red[0] * inv_n;
}

extern "C" void kernel_launch(void* const* d_in, const int* in_sizes, int n_in,
                              void* d_out, int out_size, void* d_ws, size_t ws_size,
                              hipStream_t stream) {
    const float*      features = (const float*)d_in[0];
    const long long*  labels   = (const long long*)d_in[1];
    const float*      centers  = (const float*)d_in[2];
    float*            out      = (float*)d_out;
    float*            partials = (float*)d_ws;

    const int N       = in_sizes[0] / D_DIM;       // 16384
    const int nblocks = N / ROWS_PER_BLOCK;        // 1024

    center_loss_stage1<<<nblocks, 256, 0, stream>>>(features, labels, centers,
                                                    partials);
    center_loss_stage2<<<1, 256, 0, stream>>>(partials, out, nblocks,
                                              1.0f / (float)N);
}